// NIGnetNorm_Airfoil_5188320494298
// MI455X (gfx1250) — compile-verified
//
#include <hip/hip_runtime.h>
#include <math.h>

// NIGnet forward: 64 x (2x2 linear + tanh-residual-avg) + final 2x2 + L2 normalize.
// VALU/TRANS-bound (~2.1G VALU lane-ops + 0.54G tanh vs only 64MB of HBM traffic
// = ~2.7us at 23.3 TB/s). CDNA5 paths: v_tanh_f32 hardware transcendental,
// global_load_async_to_lds_b128 (ASYNCcnt) staging + s_wait_asynccnt,
// ds_load_b128 readback, v_rsq_f32 normalize, scalar s_load weight broadcast.

#define BLOCK 256
#define PPT   4            // float4 slots per thread (2 points each -> 8 points/thread)
#define NPTS  (2 * PPT)

__device__ __forceinline__ float fast_tanh(float x) {
#if defined(__gfx1250__)
# if __has_builtin(__builtin_amdgcn_tanhf)
  return __builtin_amdgcn_tanhf(x);
# elif __has_builtin(__builtin_amdgcn_tanh_f32)
  return __builtin_amdgcn_tanh_f32(x);
# else
  float r;
  // TRANS32 op; v_nop covers the 1-op RAW hazard window (ISA 7.4).
  asm("v_tanh_f32 %0, %1\n\tv_nop" : "=v"(r) : "v"(x));
  return r;
# endif
#else
  return tanhf(x);
#endif
}

__device__ __forceinline__ float fast_rsqrt(float x) {
#if __has_builtin(__builtin_amdgcn_rsqf)
  return __builtin_amdgcn_rsqf(x);    // v_rsq_f32 (TRANS pipe)
#else
  return 1.0f / sqrtf(x);
#endif
}

__device__ __forceinline__ void stage_async(const float4* gp, const float4* lp) {
#if defined(__gfx1250__)
  const unsigned int  lds = (unsigned int)(unsigned long)lp;
  const unsigned long ga  = (unsigned long)(const void*)gp;
  asm volatile("global_load_async_to_lds_b128 %0, %1, off"
               :: "v"(lds), "v"(ga) : "memory");
#else
  *(float4*)lp = *gp;
#endif
}

// max(sqrt(n2), 1e-12) == sqrt(max(n2, 1e-24)) exactly (monotone sqrt), so the
// eps clamp folds into one v_max before a single v_rsq_f32.
__device__ __forceinline__ void finalize(float x0, float x1,
                                         float f00, float f01, float f10, float f11,
                                         float& o0, float& o1) {
  const float z0  = fmaf(f01, x1, f00 * x0);
  const float z1  = fmaf(f11, x1, f10 * x0);
  const float n2  = fmaf(z1, z1, z0 * z0);
  const float inv = fast_rsqrt(fmaxf(n2, 1e-24f));
  o0 = z0 * inv;
  o1 = z1 * inv;
}

__global__ __launch_bounds__(BLOCK) void nignet_kernel(
    const float4* __restrict__ X4,   // input points, 2 per float4
    const float*  __restrict__ Ws,   // [L,2,2] row-major
    const float*  __restrict__ Bs,   // [L,2]
    const float*  __restrict__ FW,   // [2,2] row-major
    float4*       __restrict__ out4, // output, 2 points per float4
    long n4, int nLayers)
{
  __shared__ float4 tile[BLOCK * PPT];           // 16 KB (LDS is 320 KB/WGP)
  const int  tid   = threadIdx.x;
  const long base4 = (long)blockIdx.x * (BLOCK * PPT);
  // Uniform (scalar-branch) full-tile test: avoids per-lane EXEC juggling on
  // the hot path (N = 2^22 tiles exactly; partial path kept for generality).
  const bool full  = (base4 + (long)(BLOCK * PPT)) <= n4;

  // ---- Stage input tile: global -> LDS via async DMA path (ASYNCcnt) ----
  if (full) {
#pragma unroll
    for (int i = 0; i < PPT; ++i) {
      const int slot = i * BLOCK + tid;
      stage_async(X4 + base4 + slot, &tile[slot]);
    }
  } else {
#pragma unroll
    for (int i = 0; i < PPT; ++i) {
      const int  slot = i * BLOCK + tid;
      const long g    = base4 + slot;
      if (g < n4) stage_async(X4 + g, &tile[slot]);
    }
  }
#if defined(__gfx1250__)
  asm volatile("s_wait_asynccnt 0" ::: "memory");
#endif

  // Each lane reads back exactly the slots it staged: no barrier needed.
  float x0[NPTS], x1[NPTS];
#pragma unroll
  for (int i = 0; i < PPT; ++i) {
    const float4 v = tile[i * BLOCK + tid];      // ds_load_b128
    x0[2 * i + 0] = v.x; x1[2 * i + 0] = v.y;
    x0[2 * i + 1] = v.z; x1[2 * i + 1] = v.w;
  }

  // ---- 64 layers: y = W x + b ; x = 0.5*(tanh(y)+y). 8 independent chains
  // per thread hide v_tanh_f32 (TRANS) latency; weights are wave-uniform
  // s_loads hitting the scalar cache. 8 VALU + 2 TRANS per point-layer.
  for (int l = 0; l < nLayers; ++l) {
    const float w00 = Ws[4 * l + 0], w01 = Ws[4 * l + 1];
    const float w10 = Ws[4 * l + 2], w11 = Ws[4 * l + 3];
    const float b0  = Bs[2 * l + 0], b1  = Bs[2 * l + 1];
#pragma unroll
    for (int p = 0; p < NPTS; ++p) {
      const float y0 = fmaf(w01, x1[p], fmaf(w00, x0[p], b0));
      const float y1 = fmaf(w11, x1[p], fmaf(w10, x0[p], b1));
      const float a0 = fast_tanh(y0);
      const float a1 = fast_tanh(y1);
      x0[p] = (a0 + y0) * 0.5f;
      x1[p] = (a1 + y1) * 0.5f;
    }
  }

  // ---- Final 2x2 (no bias) + row-wise L2 normalize (rsq-based) ----
  const float f00 = FW[0], f01 = FW[1], f10 = FW[2], f11 = FW[3];
  if (full) {
#pragma unroll
    for (int i = 0; i < PPT; ++i) {
      float4 o;
      finalize(x0[2*i+0], x1[2*i+0], f00, f01, f10, f11, o.x, o.y);
      finalize(x0[2*i+1], x1[2*i+1], f00, f01, f10, f11, o.z, o.w);
      out4[base4 + i * BLOCK + tid] = o;         // coalesced global_store_b128
    }
  } else {
#pragma unroll
    for (int i = 0; i < PPT; ++i) {
      const long g = base4 + (long)(i * BLOCK + tid);
      if (g >= n4) continue;
      float4 o;
      finalize(x0[2*i+0], x1[2*i+0], f00, f01, f10, f11, o.x, o.y);
      finalize(x0[2*i+1], x1[2*i+1], f00, f01, f10, f11, o.z, o.w);
      out4[g] = o;
    }
  }
}

extern "C" void kernel_launch(void* const* d_in, const int* in_sizes, int n_in,
                              void* d_out, int out_size, void* d_ws, size_t ws_size,
                              hipStream_t stream) {
  (void)n_in; (void)d_ws; (void)ws_size; (void)out_size;
  // setup_inputs order: T (unused), closed_manifold [N,2], Ws [L,2,2], bs [L,2], final_W [2,2]
  const float4* X4  = (const float4*)d_in[1];
  const float*  Ws  = (const float*)d_in[2];
  const float*  Bs  = (const float*)d_in[3];
  const float*  FW  = (const float*)d_in[4];
  float4*       out = (float4*)d_out;

  const long n4      = (long)in_sizes[1] / 4;      // float4 slots (N*2 floats / 4)
  const int  nLayers = in_sizes[2] / 4;            // L = 64
  const long perBlk  = (long)BLOCK * PPT;          // 1024 float4 slots / block
  const int  blocks  = (int)((n4 + perBlk - 1) / perBlk);

  nignet_kernel<<<blocks, BLOCK, 0, stream>>>(X4, Ws, Bs, FW, out, n4, nLayers);
}